// Encoder_13511967113592
// MI455X (gfx1250) — compile-verified
//
#include <hip/hip_runtime.h>
#include <hip/hip_bf16.h>

// ---------------------------------------------------------------------------
// GCN-VGAE encoder for MI455X (gfx1250, wave32).
//   - fp32 GEMMs on the matrix pipe via V_WMMA_F32_16X16X4_F32 (keeps
//     reference fp32 precision; GEMMs are ~13 GFLOP, far from the bottleneck)
//   - edge aggregation (the real bottleneck, ~3 GB of L2 traffic) via
//     coalesced float4 gathers + global_atomic_add_f32 scatters
// ---------------------------------------------------------------------------

typedef float v2f __attribute__((ext_vector_type(2)));
typedef float v8f __attribute__((ext_vector_type(8)));

#define N_NODES 50000
#define IN_DIM  256
#define HIDDEN  256
#define LATENT  128

// ---------------------------------------------------------------- utilities
__global__ void k_fill_f32(float* __restrict__ p, int n, float v) {
    int i = blockIdx.x * blockDim.x + threadIdx.x;
    if (i < n) p[i] = v;
}

__global__ void k_deg(const int* __restrict__ dst, float* __restrict__ deg, int E) {
    int e = blockIdx.x * blockDim.x + threadIdx.x;
    if (e < E) atomicAdd(&deg[dst[e]], 1.0f);
}

__global__ void k_rsqrt_inplace(float* __restrict__ p, int n) {
    int i = blockIdx.x * blockDim.x + threadIdx.x;
    if (i < n) p[i] = rsqrtf(p[i]);
}

__global__ void k_edge_norm(const int* __restrict__ src, const int* __restrict__ dst,
                            const float* __restrict__ dinv, float* __restrict__ norm, int E) {
    int e = blockIdx.x * blockDim.x + threadIdx.x;
    if (e < E) norm[e] = dinv[src[e]] * dinv[dst[e]];
}

// ------------------------------------------------------------- fp32 WMMA GEMM
// C[M,N] = A[M,K] * B[K,N], row-major, all fp32.
// One wave computes a 16x64 tile: 4 accumulators reuse each A fragment 4x and
// give 4 independent WMMA dependency chains per K-step.
// Fragment layout (ISA 7.12.2, 32-bit operands):
//   A: lane (m = lane&15, kg = lane>>4) holds A[m][k0+2kg], A[m][k0+2kg+1]
//   B: lane (n = lane&15, kg = lane>>4) holds B[k0+2kg][n], B[k0+2kg+1][n]
//   C/D: VGPR r, lane -> C[16*mt + r + 8*(lane>>4)][n]
template <int NT>
__global__ __launch_bounds__(128) void k_gemm_wmma_f32(
    const float* __restrict__ A, const float* __restrict__ B, float* __restrict__ C,
    int M, int N, int K) {
    const int lane  = threadIdx.x & 31;
    const int wave  = threadIdx.x >> 5;
    const int mtile = blockIdx.y * 4 + wave;            // 4 waves / block
    if (mtile * 16 >= M) return;                        // whole-wave guard (EXEC all-1s)
    const int ln    = lane & 15;
    const int kg    = lane >> 4;                        // K-half selector
    const int nbase = blockIdx.x * (16 * NT);
    const long arow = (long)(mtile * 16 + ln) * K;

    v8f acc[NT] = {};
    for (int k0 = 0; k0 < K; k0 += 4) {
        const int ka = k0 + 2 * kg;
        v2f a = *(const v2f*)(A + arow + ka);           // 8B load, coalesced over lanes
#pragma unroll
        for (int t = 0; t < NT; ++t) {
            const int n = nbase + t * 16 + ln;
            v2f b;
            b.x = B[(long)ka * N + n];
            b.y = B[(long)(ka + 1) * N + n];
            acc[t] = __builtin_amdgcn_wmma_f32_16x16x4_f32(
                /*neg_a=*/false, a, /*neg_b=*/false, b,
                /*c_mod=*/(short)0, acc[t], /*reuse_a=*/false, /*reuse_b=*/false);
        }
    }
#pragma unroll
    for (int t = 0; t < NT; ++t) {
        const int n = nbase + t * 16 + ln;
#pragma unroll
        for (int r = 0; r < 8; ++r) {
            C[(long)(mtile * 16 + r + 8 * kg) * N + n] = acc[t][r];
        }
    }
}

// ----------------------------------------------- aggregation (self loop+bias)
// agg[i][d] = h[i][d] * dinv[i]^2 + bias[d];  D = 1<<shift (256 or 128)
__global__ void k_init_agg(const float* __restrict__ h, const float* __restrict__ dinv,
                           const float* __restrict__ bias, float* __restrict__ agg,
                           long total, int shift) {
    long tid = (long)blockIdx.x * blockDim.x + threadIdx.x;
    if (tid >= total) return;
    const int  d  = (int)(tid & ((1 << shift) - 1));
    const int  i  = (int)(tid >> shift);
    const float di = dinv[i];
    agg[tid] = fmaf(h[tid], di * di, bias[d]);
}

// ---------------------------------------------------------- edge scatter-add
// One thread per (edge, 4-dim chunk): coalesced float4 gather of h[src],
// 4x global_atomic_add_f32 into agg[dst]. shift = log2(D/4).
__global__ void k_scatter(const float* __restrict__ h, const int* __restrict__ src,
                          const int* __restrict__ dst, const float* __restrict__ norm,
                          float* __restrict__ agg, int E, int shift) {
    long tid = (long)blockIdx.x * blockDim.x + threadIdx.x;
    const int e = (int)(tid >> shift);
    if (e >= E) return;
    const int c   = (int)(tid & ((1 << shift) - 1)) << 2;     // dim offset (floats)
    const float nm = norm[e];
    const long so = ((long)src[e] << (shift + 2)) + c;        // node*D + c
    const long dofs = ((long)dst[e] << (shift + 2)) + c;
    const float4 hv = *(const float4*)(h + so);
    atomicAdd(agg + dofs + 0, hv.x * nm);
    atomicAdd(agg + dofs + 1, hv.y * nm);
    atomicAdd(agg + dofs + 2, hv.z * nm);
    atomicAdd(agg + dofs + 3, hv.w * nm);
}

__global__ void k_relu_inplace(float* __restrict__ p, long n) {
    long i = (long)blockIdx.x * blockDim.x + threadIdx.x;
    if (i < n) p[i] = fmaxf(p[i], 0.0f);
}

// ---------------------------------------------------------------------------
extern "C" void kernel_launch(void* const* d_in, const int* in_sizes, int n_in,
                              void* d_out, int out_size, void* d_ws, size_t ws_size,
                              hipStream_t stream) {
    const float* x   = (const float*)d_in[0];
    const int*   ei  = (const int*)d_in[1];
    const float* W1  = (const float*)d_in[2];
    const float* b1  = (const float*)d_in[3];
    const float* Wmu = (const float*)d_in[4];
    const float* bmu = (const float*)d_in[5];
    const float* Wlv = (const float*)d_in[6];
    const float* blv = (const float*)d_in[7];
    float* out = (float*)d_out;

    const int E = in_sizes[1] / 2;
    const int* src = ei;
    const int* dst = ei + E;

    // ---- workspace carve-out (256B aligned) : ~106 MB total
    char*  ws  = (char*)d_ws;
    size_t off = 0;
    auto carve = [&](size_t bytes) -> void* {
        void* p = ws + off;
        off = (off + bytes + 255) & ~(size_t)255;
        return p;
    };
    float* dinv = (float*)carve((size_t)N_NODES * 4);
    float* norm = (float*)carve((size_t)E * 4);
    float* h    = (float*)carve((size_t)N_NODES * HIDDEN * 4);
    float* agg  = (float*)carve((size_t)N_NODES * HIDDEN * 4);
    float* hmu  = h;                                   // h is dead after relu(agg)
    float* hlv  = h + (size_t)N_NODES * LATENT;
    float* out_mu = out;
    float* out_lv = out + (size_t)N_NODES * LATENT;

    const int T = 256;
    const int MT = (N_NODES / 16 + 3) / 4;             // 3125 m-tiles, 4 waves/block

    // ---- degrees, dinv, per-edge norm (reused by all 3 aggregations)
    k_fill_f32<<<(N_NODES + T - 1) / T, T, 0, stream>>>(dinv, N_NODES, 1.0f); // self loop
    k_deg<<<(E + T - 1) / T, T, 0, stream>>>(dst, dinv, E);
    k_rsqrt_inplace<<<(N_NODES + T - 1) / T, T, 0, stream>>>(dinv, N_NODES);
    k_edge_norm<<<(E + T - 1) / T, T, 0, stream>>>(src, dst, dinv, norm, E);

    // ---- layer 1: h = x @ W1 ; agg = scatter + self + b1 ; z = relu(agg)
    {
        dim3 g(HIDDEN / 64, MT);
        k_gemm_wmma_f32<4><<<g, 128, 0, stream>>>(x, W1, h, N_NODES, HIDDEN, IN_DIM);
        long tot = (long)N_NODES * HIDDEN;
        k_init_agg<<<(int)((tot + T - 1) / T), T, 0, stream>>>(h, dinv, b1, agg, tot, 8);
        long st = (long)E * (HIDDEN / 4);
        k_scatter<<<(int)((st + T - 1) / T), T, 0, stream>>>(h, src, dst, norm, agg, E, 6);
        k_relu_inplace<<<(int)((tot + T - 1) / T), T, 0, stream>>>(agg, tot);
    }

    // ---- layer 2: mu / logvar (aggregate straight into d_out)
    {
        dim3 g(LATENT / 64, MT);
        k_gemm_wmma_f32<4><<<g, 128, 0, stream>>>(agg, Wmu, hmu, N_NODES, LATENT, HIDDEN);
        k_gemm_wmma_f32<4><<<g, 128, 0, stream>>>(agg, Wlv, hlv, N_NODES, LATENT, HIDDEN);
        long tot = (long)N_NODES * LATENT;
        k_init_agg<<<(int)((tot + T - 1) / T), T, 0, stream>>>(hmu, dinv, bmu, out_mu, tot, 7);
        k_init_agg<<<(int)((tot + T - 1) / T), T, 0, stream>>>(hlv, dinv, blv, out_lv, tot, 7);
        long st = (long)E * (LATENT / 4);
        k_scatter<<<(int)((st + T - 1) / T), T, 0, stream>>>(hmu, src, dst, norm, out_mu, E, 5);
        k_scatter<<<(int)((st + T - 1) / T), T, 0, stream>>>(hlv, src, dst, norm, out_lv, E, 5);
    }
}